// BoundaryLoss_90804198572783
// MI455X (gfx1250) — compile-verified
//
#include <hip/hip_runtime.h>
#include <hip/hip_bf16.h>

#define HH 256
#define WW 256
#define NPIX (HH * WW)
#define BIGF 1.0e6f

typedef float v2f __attribute__((ext_vector_type(2)));
typedef float v8f __attribute__((ext_vector_type(8)));

__device__ __forceinline__ float fast_sigmoid(float x) {
    // v_exp_f32 + v_rcp_f32 (avoids the IEEE div_fixup sequence)
    return __builtin_amdgcn_rcpf(1.0f + __expf(-x));
}

// ---------------- init: zero per-image max ----------------
__global__ void bl_init(unsigned* __restrict__ dmax, int B) {
    int t = threadIdx.x;
    if (t < B) dmax[t] = 0u;
}

// ---------------- phase 1: binarize, erode (bitmask), vertical EDT scan ----
// writes s[b][i][j] = g(i,j)^2 + j^2 ; flags[b] = has_fg
__global__ void bl_phase1(const float* __restrict__ target,
                          float* __restrict__ s,
                          unsigned* __restrict__ flags) {
    const int b = blockIdx.x;
    const int t = threadIdx.x;          // 256 threads = 8 waves (wave32)
    const int lane = t & 31, wave = t >> 5;

    __shared__ unsigned binm[HH][8];    // binary mask, bit j of word w = col 32w+j
    __shared__ unsigned aux[HH][8];     // h-eroded, then boundary mask
    __shared__ int cntBound, cntBin, useBound;

    if (t == 0) { cntBound = 0; cntBin = 0; }

    const float* timg = target + (size_t)b * NPIX;

    // build bitmask rows via wave32 ballot (coalesced 128B loads per wave)
    for (int r = wave; r < HH; r += 8) {
        for (int c = 0; c < 8; ++c) {
            float v = timg[r * WW + c * 32 + lane];
            unsigned m = __builtin_amdgcn_ballot_w32(v > 0.5f);
            if (lane == 0) binm[r][c] = m;
        }
    }
    __syncthreads();

    // horizontal 3-wide min (erosion, out-of-bounds never erodes -> border=1)
    for (int w = 0; w < 8; ++w) {
        unsigned bcur = binm[t][w];
        unsigned lft = (bcur << 1) | (w > 0 ? (binm[t][w - 1] >> 31) : 1u);
        unsigned rgt = (bcur >> 1) | ((w < 7 ? (binm[t][w + 1] & 1u) : 1u) << 31);
        aux[t][w] = bcur & lft & rgt;
    }
    __syncthreads();

    // vertical 3-tall min -> eroded; boundaries = bin & ~eroded; count bits
    unsigned boundw[8];
    int myB = 0, myN = 0;
    for (int w = 0; w < 8; ++w) {
        unsigned up = (t > 0)   ? aux[t - 1][w] : 0xFFFFFFFFu;
        unsigned dn = (t < 255) ? aux[t + 1][w] : 0xFFFFFFFFu;
        unsigned e  = aux[t][w] & up & dn;
        unsigned bo = binm[t][w] & ~e;
        boundw[w] = bo;
        myB += __popc(bo);
        myN += __popc(binm[t][w]);
    }
    __syncthreads();                    // all reads of h done before overwrite
    for (int w = 0; w < 8; ++w) aux[t][w] = boundw[w];
    if (myB) atomicAdd(&cntBound, myB);
    if (myN) atomicAdd(&cntBin, myN);
    __syncthreads();
    if (t == 0) {
        useBound = (cntBound > 0) ? 1 : 0;
        flags[b] = (cntBin > 0) ? 1u : 0u;
    }
    __syncthreads();

    const unsigned (*fm)[8] = useBound ? aux : binm;  // uniform select

    // vertical nearest-feature distance: thread t = column j
    const int j = t;
    float* simg = s + (size_t)b * NPIX;
    float c = BIGF;
    for (int i = HH - 1; i >= 0; --i) {               // bottom-up scan
        unsigned word = fm[i][j >> 5];                // LDS broadcast read
        bool f = (word >> (j & 31)) & 1u;
        c = f ? 0.0f : (c + 1.0f);
        simg[i * WW + j] = c;                         // coalesced store
    }
    c = BIGF;
    const float j2 = (float)(j * j);
    for (int i = 0; i < HH; ++i) {                    // top-down scan + combine
        unsigned word = fm[i][j >> 5];
        bool f = (word >> (j & 31)) & 1u;
        c = f ? 0.0f : (c + 1.0f);
        float g = fminf(simg[i * WW + j], c);
        simg[i * WW + j] = fmaf(g, g, j2);            // s = g^2 + j^2
    }
}

// ---------------- phase 2: per-row lower envelope (min-plus), in-place -----
// dist[b][i][j] = sqrt(min_{j'} (s[j'] - 2 j j') + j^2) ; also per-image max
__global__ void bl_phase2(float* __restrict__ s, unsigned* __restrict__ dmax) {
    const int row = blockIdx.x;         // b*256 + i
    const int b = row >> 8;
    const int t = threadIdx.x;          // column j
    __shared__ float sl[WW];
    __shared__ unsigned blkmax;

    float* srow = s + (size_t)row * WW;
    sl[t] = srow[t];
    if (t == 0) blkmax = 0u;
    __syncthreads();

    const float n2j = -2.0f * (float)t;
    float m = 3.0e38f;
#pragma unroll 8
    for (int q = 0; q < WW; ++q)
        m = fminf(m, fmaf(n2j, (float)q, sl[q]));     // s[j'] - 2 j j'

    float d = sqrtf(fmaxf(m + (float)(t * t), 0.0f));
    srow[t] = d;                                      // overwrite s with dist
    atomicMax(&blkmax, __float_as_uint(d));           // d >= 0: bit order ok
    __syncthreads();
    if (t == 0) atomicMax(&dmax[b], blkmax);
}

// ---------------- phase 3: WMMA ones-reduction of dist*|sigmoid(x)-t| ------
__global__ void bl_loss(const float* __restrict__ dist,
                        const float* __restrict__ logits,
                        const float* __restrict__ target,
                        const unsigned* __restrict__ dmax,
                        const unsigned* __restrict__ flags,
                        float* __restrict__ lossPer) {
    const int b = blockIdx.x;
    const int t = threadIdx.x;          // 256 threads = 8 waves
    const int lane = t & 31, wave = t >> 5;
    __shared__ float part[256];

    const float m = __uint_as_float(dmax[b]);
    const float inv = (m > 0.0f) ? (1.0f / (m + 1e-8f)) : 1.0f;

    const float2* dp = (const float2*)(dist   + (size_t)b * NPIX);
    const float2* lp = (const float2*)(logits + (size_t)b * NPIX);
    const float2* tp = (const float2*)(target + (size_t)b * NPIX);

    v8f acc = {0.f, 0.f, 0.f, 0.f, 0.f, 0.f, 0.f, 0.f};
    const v2f bones = {1.0f, 1.0f};

    // 65536 elems / 8 waves = 8192 per wave = 128 WMMAs x 64 elems
    for (int k = 0; k < 128; ++k) {
        int idx = (k * 8 + wave) * 32 + lane;         // float2 index
        float2 dv = dp[idx];
        float2 lv = lp[idx];
        float2 tv = tp[idx];
        float p0 = (dv.x * inv) * fabsf(fast_sigmoid(lv.x) - tv.x);
        float p1 = (dv.y * inv) * fabsf(fast_sigmoid(lv.y) - tv.y);
        v2f a = {p0, p1};
        // D[m][n] += sum_k A[m][k] * 1  ->  sum(D) = 16 * sum(A)
        acc = __builtin_amdgcn_wmma_f32_16x16x4_f32(
            false, a, false, bones, (short)0, acc, false, false);
    }
    part[t] = acc[0] + acc[1] + acc[2] + acc[3] +
              acc[4] + acc[5] + acc[6] + acc[7];
    __syncthreads();
    if (t == 0) {
        float tot = 0.f;
        for (int q = 0; q < 256; ++q) tot += part[q];
        tot *= (1.0f / 16.0f);                        // undo ones replication
        lossPer[b] = flags[b] ? (tot / (float)NPIX) : 0.0f;
    }
}

// ---------------- final: average over batch --------------------------------
__global__ void bl_final(const float* __restrict__ lossPer, int B,
                         float* __restrict__ out) {
    if (threadIdx.x == 0) {
        float ssum = 0.f;
        for (int q = 0; q < B; ++q) ssum += lossPer[q];
        out[0] = ssum / (float)B;
    }
}

extern "C" void kernel_launch(void* const* d_in, const int* in_sizes, int n_in,
                              void* d_out, int out_size, void* d_ws, size_t ws_size,
                              hipStream_t stream) {
    const float* logits = (const float*)d_in[0];
    const float* target = (const float*)d_in[1];
    float* out = (float*)d_out;

    const int B = in_sizes[0] / NPIX;   // 8 for the reference shapes

    float*    s       = (float*)d_ws;                 // B*65536 floats (dist in-place)
    unsigned* dmax    = (unsigned*)(s + (size_t)B * NPIX);
    unsigned* flags   = dmax + B;
    float*    lossPer = (float*)(flags + B);

    bl_init  <<<1, 32, 0, stream>>>(dmax, B);
    bl_phase1<<<B, 256, 0, stream>>>(target, s, flags);
    bl_phase2<<<B * HH, 256, 0, stream>>>(s, dmax);
    bl_loss  <<<B, 256, 0, stream>>>(s, logits, target, dmax, flags, lossPer);
    bl_final <<<1, 32, 0, stream>>>(lossPer, B, out);
}